// TemporalGAT_59072980189186
// MI455X (gfx1250) — compile-verified
//
#include <hip/hip_runtime.h>
#include <hip/hip_bf16.h>

typedef __attribute__((ext_vector_type(2))) float v2f;
typedef __attribute__((ext_vector_type(8))) float v8f;

#define LRELU_SLOPE 0.5f

__device__ __forceinline__ float lrelu(float x) { return x > 0.f ? x : LRELU_SLOPE * x; }

// ---------------------------------------------------------------------------
// GEMM C[M,N] = A[M,K] @ B[K,N] (all row-major, f32) using V_WMMA_F32_16X16X4_F32.
// One wave computes a 16x64 output tile. grid = (M/16, N/64, heads).
// A-operand layout: lane l holds row (l&15), K = k0 + 2*(l>>4) + {0,1}
// B-operand layout: lane l holds col (l&15), K = k0 + 2*(l>>4) + {0,1}
// C/D layout: vgpr v -> row v + 8*(l>>4), col (l&15)
// ---------------------------------------------------------------------------
__global__ __launch_bounds__(32) void gemm_f32_wmma(
    const float* __restrict__ A, const float* __restrict__ B, float* __restrict__ C,
    int K, int lda, int ldb, int ldc, size_t b_hstride, size_t c_hstride)
{
  const int lane = threadIdx.x & 31;
  const int half = lane >> 4;
  const int q    = lane & 15;
  const float* Bh = B + (size_t)blockIdx.z * b_hstride;
  float*       Ch = C + (size_t)blockIdx.z * c_hstride;
  const int r0 = blockIdx.x * 16;
  const int n0 = blockIdx.y * 64;
  const float* arow = A + (size_t)(r0 + q) * lda;

  v8f acc[4];
#pragma unroll
  for (int t = 0; t < 4; ++t)
#pragma unroll
    for (int v = 0; v < 8; ++v) acc[t][v] = 0.f;

#pragma unroll 2
  for (int k0 = 0; k0 < K; k0 += 4) {
    const int kk = k0 + 2 * half;
    float2 af = *(const float2*)(arow + kk);
    v2f av; av.x = af.x; av.y = af.y;
    const float* b0 = Bh + (size_t)kk * ldb;
    const float* b1 = b0 + ldb;
#pragma unroll
    for (int t = 0; t < 4; ++t) {
      const int col = n0 + t * 16 + q;
      v2f bv; bv.x = b0[col]; bv.y = b1[col];
      acc[t] = __builtin_amdgcn_wmma_f32_16x16x4_f32(
          false, av, false, bv, (short)0, acc[t], false, false);
    }
  }
#pragma unroll
  for (int t = 0; t < 4; ++t)
#pragma unroll
    for (int v = 0; v < 8; ++v)
      Ch[(size_t)(r0 + v + 8 * half) * ldc + n0 + t * 16 + q] = acc[t][v];
}

// ---------------------------------------------------------------------------
// f1[i] = Wh[i,:] . a[:F]   f2[i] = Wh[i,:] . a[F:2F]   (per head via blockIdx.y)
// ---------------------------------------------------------------------------
__global__ __launch_bounds__(256) void fvec_kernel(
    const float* __restrict__ Wh, const float* __restrict__ a,
    float* __restrict__ f1, float* __restrict__ f2, int n, int F)
{
  const int h = blockIdx.y;
  const int r = blockIdx.x * blockDim.x + threadIdx.x;
  if (r >= n) return;
  const float* row = Wh + ((size_t)h * n + r) * F;
  const float* ah  = a + (size_t)h * 2 * F;
  float s1 = 0.f, s2 = 0.f;
  for (int c = 0; c < F; ++c) { float w = row[c]; s1 += w * ah[c]; s2 += w * ah[F + c]; }
  f1[h * n + r] = s1;
  f2[h * n + r] = s2;
}

// ---------------------------------------------------------------------------
// Per-row online softmax stats over masked scores, all heads fused so the 67MB
// adjacency is scanned once. One wave per row; lanes stride j with int4 loads.
// Outputs: mout = row max, isout = 1/row_sum.
// ---------------------------------------------------------------------------
#define STAT_UPD(MKF, FVF)                                                     \
  if ((MKF) > 0) {                                                             \
    float e_ = lrelu(f1v[h] + (FVF));                                          \
    if (e_ > m[h]) { s[h] = s[h] * __expf(m[h] - e_) + 1.f; m[h] = e_; }       \
    else s[h] += __expf(e_ - m[h]);                                            \
  }

__global__ __launch_bounds__(256) void rowstats_kernel(
    const int* __restrict__ adj, const float* __restrict__ f1,
    const float* __restrict__ f2, float* __restrict__ mout,
    float* __restrict__ isout, int nh, int n)
{
  const int row  = blockIdx.x * (blockDim.x >> 5) + (threadIdx.x >> 5);
  const int lane = threadIdx.x & 31;
  if (row >= n) return;
  const int* arow = adj + (size_t)row * n;

  float f1v[4], m[4], s[4];
  for (int h = 0; h < 4; ++h) {
    m[h] = -1e30f; s[h] = 0.f;
    f1v[h] = (h < nh) ? f1[h * n + row] : 0.f;
  }
  const int iters = n / 128;
  for (int c = 0; c < iters; ++c) {
    const int j = c * 128 + lane * 4;
    int4 mk = *(const int4*)(arow + j);
    for (int h = 0; h < nh; ++h) {
      float4 fv = *(const float4*)(f2 + (size_t)h * n + j);
      STAT_UPD(mk.x, fv.x)
      STAT_UPD(mk.y, fv.y)
      STAT_UPD(mk.z, fv.z)
      STAT_UPD(mk.w, fv.w)
    }
  }
  for (int h = 0; h < nh; ++h) {
    for (int off = 16; off > 0; off >>= 1) {
      float om = __shfl_xor(m[h], off, 32);
      float os = __shfl_xor(s[h], off, 32);
      float nm = fmaxf(m[h], om);
      s[h] = s[h] * __expf(m[h] - nm) + os * __expf(om - nm);
      m[h] = nm;
    }
    if (lane == 0) {
      mout[h * n + row]  = m[h];
      isout[h * n + row] = (s[h] > 0.f) ? 1.f / s[h] : 0.f;
    }
  }
}

// ---------------------------------------------------------------------------
// Fused attention + aggregation: out[16 rows, FT*16 cols] = softmax(e) @ V,
// un-normalized probabilities exp(e - rowmax) generated on the fly as WMMA
// A-operands (no [N,N] matrix); the 1/rowsum normalization is linear, so it is
// applied once to the reduced accumulator instead of per edge (shortens the
// exp -> WMMA dependency chain). 8 waves per block split the j loop
// (wave-uniform => EXEC all-1s for WMMA), partials combined via LDS.
// ---------------------------------------------------------------------------
template <int FT>
__global__ __launch_bounds__(256) void attn_kernel(
    const float* __restrict__ V, const float* __restrict__ f1,
    const float* __restrict__ f2, const float* __restrict__ mrow,
    const float* __restrict__ isrow, const int* __restrict__ adj,
    float* __restrict__ out, int n, int ldv, int ldo, int head_ostride,
    size_t v_hstride, int f_hstride, int apply_elu)
{
  __shared__ float red[256 * FT * 8];
  const int tid  = threadIdx.x;
  const int w    = tid >> 5;
  const int lane = tid & 31;
  const int half = lane >> 4;
  const int q    = lane & 15;
  const int h    = blockIdx.z;

  const float* Vh  = V + (size_t)h * v_hstride;
  const float* f1h = f1 + (size_t)h * f_hstride;
  const float* f2h = f2 + (size_t)h * f_hstride;
  const float* mh  = mrow + (size_t)h * f_hstride;
  const float* ih  = isrow + (size_t)h * f_hstride;

  const int r0    = blockIdx.x * 16;
  const int myrow = r0 + q;
  const float fl = f1h[myrow];
  const float mr = mh[myrow];
  const int* arow = adj + (size_t)myrow * n;

  v8f acc[FT];
#pragma unroll
  for (int t = 0; t < FT; ++t)
#pragma unroll
    for (int v = 0; v < 8; ++v) acc[t][v] = 0.f;

#pragma unroll 2
  for (int j0 = w * 4; j0 < n; j0 += 32) {
    const int jk = j0 + 2 * half;
    float2 fj = *(const float2*)(f2h + jk);
    int2   mk = *(const int2*)(arow + jk);
    float e0 = lrelu(fl + fj.x);
    float e1 = lrelu(fl + fj.y);
    float p0 = (mk.x > 0) ? __expf(e0 - mr) : 0.f;
    float p1 = (mk.y > 0) ? __expf(e1 - mr) : 0.f;
    v2f pa; pa.x = p0; pa.y = p1;
    const float* v0 = Vh + (size_t)jk * ldv;
    const float* v1 = v0 + ldv;
#pragma unroll
    for (int t = 0; t < FT; ++t) {
      const int col = t * 16 + q;
      v2f bv; bv.x = v0[col]; bv.y = v1[col];
      acc[t] = __builtin_amdgcn_wmma_f32_16x16x4_f32(
          false, pa, false, bv, (short)0, acc[t], false, false);
    }
  }

  float* myred = red + (size_t)tid * (FT * 8);
#pragma unroll
  for (int t = 0; t < FT; ++t)
#pragma unroll
    for (int v = 0; v < 8; ++v) myred[t * 8 + v] = acc[t][v];
  __syncthreads();

  if (tid < 32) {
    for (int ww = 1; ww < 8; ++ww) {
      const float* o = red + (size_t)(ww * 32 + lane) * (FT * 8);
#pragma unroll
      for (int t = 0; t < FT; ++t)
#pragma unroll
        for (int v = 0; v < 8; ++v) acc[t][v] += o[t * 8 + v];
    }
    // Deferred softmax normalization: scale reduced tile rows by 1/rowsum.
    float isv[8];
#pragma unroll
    for (int v = 0; v < 8; ++v) isv[v] = ih[r0 + v + 8 * half];
    const int ooff = h * head_ostride;
#pragma unroll
    for (int t = 0; t < FT; ++t) {
#pragma unroll
      for (int v = 0; v < 8; ++v) {
        const int row = r0 + v + 8 * half;
        const int col = ooff + t * 16 + q;
        float val = acc[t][v] * isv[v];
        if (apply_elu) val = val > 0.f ? val : __expf(val) - 1.f;
        out[(size_t)row * ldo + col] = val;
      }
    }
  }
}

// ---------------------------------------------------------------------------
extern "C" void kernel_launch(void* const* d_in, const int* in_sizes, int n_in,
                              void* d_out, int out_size, void* d_ws, size_t ws_size,
                              hipStream_t stream)
{
  const float* x       = (const float*)d_in[0];  // [4096, 256]
  const int*   adj     = (const int*)  d_in[1];  // [4096, 4096]
  const float* W_heads = (const float*)d_in[2];  // [4, 256, 64]
  const float* a_heads = (const float*)d_in[3];  // [4, 128]
  const float* W_out   = (const float*)d_in[4];  // [256, 128]
  const float* a_out   = (const float*)d_in[5];  // [256]
  float* out = (float*)d_out;                    // [4096, 128]

  const int n = 4096, IN = 256, HID = 64, NH = 4, OD = 128;

  float* ws = (float*)d_ws;
  size_t off = 0;
  auto alloc = [&](size_t cnt) {
    float* p = ws + off;
    off += (cnt + 63) & ~(size_t)63;
    return p;
  };
  float* Wh1 = alloc((size_t)NH * n * HID);  // per-head x@W
  float* f1h = alloc((size_t)NH * n);
  float* f2h = alloc((size_t)NH * n);
  float* m1  = alloc((size_t)NH * n);
  float* is1 = alloc((size_t)NH * n);
  float* h1  = alloc((size_t)n * (NH * HID)); // concat layer-1 output
  float* Wh2 = alloc((size_t)n * OD);
  float* f1o = alloc(n);
  float* f2o = alloc(n);
  float* m2  = alloc(n);
  float* is2 = alloc(n);
  (void)in_sizes; (void)n_in; (void)out_size; (void)ws_size;

  // --- Layer 1 ---
  gemm_f32_wmma<<<dim3(n / 16, 1, NH), 32, 0, stream>>>(
      x, W_heads, Wh1, IN, IN, HID, HID, (size_t)IN * HID, (size_t)n * HID);
  fvec_kernel<<<dim3(n / 256, NH), 256, 0, stream>>>(Wh1, a_heads, f1h, f2h, n, HID);
  rowstats_kernel<<<dim3(n / 8), 256, 0, stream>>>(adj, f1h, f2h, m1, is1, NH, n);
  attn_kernel<4><<<dim3(n / 16, 1, NH), 256, 0, stream>>>(
      Wh1, f1h, f2h, m1, is1, adj, h1,
      n, HID, NH * HID, HID, (size_t)n * HID, n, /*elu=*/1);

  // --- Layer 2 ---
  gemm_f32_wmma<<<dim3(n / 16, OD / 64, 1), 32, 0, stream>>>(
      h1, W_out, Wh2, NH * HID, NH * HID, OD, OD, 0, 0);
  fvec_kernel<<<dim3(n / 256, 1), 256, 0, stream>>>(Wh2, a_out, f1o, f2o, n, OD);
  rowstats_kernel<<<dim3(n / 8), 256, 0, stream>>>(adj, f1o, f2o, m2, is2, 1, n);
  attn_kernel<8><<<dim3(n / 16, 1, 1), 256, 0, stream>>>(
      Wh2, f1o, f2o, m2, is2, adj, out,
      n, OD, OD, 0, 0, 0, /*elu=*/1);
}